// GPTQCompressedLayer_45397804318855
// MI455X (gfx1250) — compile-verified
//
#include <hip/hip_runtime.h>
#include <hip/hip_bf16.h>

// GPTQ 4-bit 2:4-sparse GEMM for MI455X (gfx1250).
// Pass 1: x fp32 -> fp16            (16 MB, L2-resident)
// Pass 2: dequant 4-bit -> fp16     ([N, KC] = 45 MB, L2-resident; LDS transpose
//                                    for coalesced reads AND writes)
// Pass 3: V_SWMMAC_F32_16X16X64_F16 GEMM, 64n x 64m register tile per wave,
//         __launch_bounds__(32,1) so the allocator can hold 16 f32 accumulators
//         (128 VGPRs) + operands in registers without scratch spills.

#define MDIM  2048
#define KDIM  4096
#define NDIM  11008
#define KCDIM 2048            // KDIM / 2 (2:4 compressed)
#define GROUPSZ 128

typedef __attribute__((ext_vector_type(4)))  _Float16 v4h;
typedef __attribute__((ext_vector_type(8)))  _Float16 v8h;
typedef __attribute__((ext_vector_type(16))) _Float16 v16h;
typedef __attribute__((ext_vector_type(32))) _Float16 v32h;
typedef __attribute__((ext_vector_type(8)))  float    v8f;

// D = A(16x64 sparse f16, stored 16x32) * B(64x16 f16) + C(f32), idx = 2-bit pairs
#define SWMMAC_F16(A, B, C, IDX) \
  __builtin_amdgcn_swmmac_f32_16x16x64_f16(false, (A), false, (B), (C), (IDX), false, false)

static __device__ __forceinline__ v16h cat8(v8h a, v8h b) {
  return __builtin_shufflevector(a, b, 0,1,2,3,4,5,6,7,8,9,10,11,12,13,14,15);
}
static __device__ __forceinline__ v32h cat16(v16h a, v16h b) {
  return __builtin_shufflevector(a, b, 0,1,2,3,4,5,6,7,8,9,10,11,12,13,14,15,
                                 16,17,18,19,20,21,22,23,24,25,26,27,28,29,30,31);
}

// ---------------- Pass 1: x fp32 -> fp16 ----------------
__global__ void xcvt_kernel(const float* __restrict__ x, _Float16* __restrict__ xh) {
  size_t i = ((size_t)blockIdx.x * blockDim.x + threadIdx.x) * 4;
  float4 v = *(const float4*)(x + i);
  v4h h = { (_Float16)v.x, (_Float16)v.y, (_Float16)v.z, (_Float16)v.w };
  *(v4h*)(xh + i) = h;
}

// ---------------- Pass 2: dequantize compressed weights -> wh[N, KC] f16 ----
// Block tile: 32 n x 8 qweight-words (64 kc). Phase 1 reads qweight coalesced
// (consecutive n across lanes) and dequantizes into an LDS tile; phase 2
// re-reads the tile transposed and writes wh[N, KC] coalesced (128B runs).
__global__ void dequant_kernel(const int* __restrict__ qweight,   // [KC/8, N]
                               const int* __restrict__ g_idx,     // [KC]
                               const int* __restrict__ qzeros,    // [K/GROUP, N/8]
                               const float* __restrict__ scales,  // [K/GROUP, N]
                               _Float16* __restrict__ wh)         // [N, KC]
{
  __shared__ _Float16 tile[32][64];                // 4 KB
  const int t    = threadIdx.x;                    // 0..255
  const int n0   = blockIdx.x * 32;
  const int kcw0 = blockIdx.y * 8;

  // phase 1: t = kcwSub*32 + nSub  -> coalesced qweight/scales reads
  {
    const int nSub = t & 31, kcwSub = t >> 5;
    const int n = n0 + nSub, kcw = kcw0 + kcwSub;
    const int word = qweight[(size_t)kcw * NDIM + n];
    const int g    = g_idx[kcw * 8];               // uniform over the 8-kc word
    const int zw   = qzeros[(size_t)g * (NDIM / 8) + (n >> 3)];
    const int z    = ((zw >> (4 * (n & 7))) & 0xF) + 1;
    const float s  = scales[(size_t)g * NDIM + n];
#pragma unroll
    for (int j = 0; j < 8; ++j)
      tile[nSub][kcwSub * 8 + j] = (_Float16)(s * (float)(((word >> (4 * j)) & 0xF) - z));
  }
  __syncthreads();
  // phase 2: t = nSub2*8 + kcwSub2 -> coalesced 128B wh writes per 8-lane group
  {
    const int kcwSub2 = t & 7, nSub2 = t >> 3;
    v8h w = *(const v8h*)&tile[nSub2][kcwSub2 * 8];
    *(v8h*)(wh + (size_t)(n0 + nSub2) * KCDIM + (size_t)(kcw0 + kcwSub2) * 8) = w;
  }
}

// ---------------- Pass 3: SWMMAC GEMM ----------------
// grid = (N/64, M/64), block = 32 threads (1 wave). out[n, m] row-major.
// Per 64-k step: 4 A tiles (v16h) + 4 index words + 4 B tiles (v32h),
// B loaded one at a time and consumed by 4 swmmac to bound VGPR liveness.
// min-waves-per-EU = 1: this loop hides latency with 16 independent swmmac
// per step, not wave switching, so give the allocator the full VGPR file.
__global__ void __launch_bounds__(32, 1)
gemm_swmmac_kernel(const _Float16* __restrict__ wh,  // [N, KC]
                   const int* __restrict__ meta,     // [N, KC/8]
                   const _Float16* __restrict__ xh,  // [M, K]
                   float* __restrict__ out)          // [N, M]
{
  const int lane = threadIdx.x;
  const int lo = lane & 15;
  const int hi = lane >> 4;                        // 0 or 1 (half-wave group)
  const int n0 = blockIdx.x * 64;
  const int m0 = blockIdx.y * 64;

  v8f acc[4][4] = {};

  // A: lane row = n0 + 16*i + lo; halves {0..7}@kc0+8*hi, {8..15}@kc0+16+8*hi
  const _Float16* pa[4];
  const int*      pm[4];
  const _Float16* pb[4];
#pragma unroll
  for (int i = 0; i < 4; ++i) {
    pa[i] = wh   + (size_t)(n0 + 16 * i + lo) * KCDIM     + 8 * hi;
    pm[i] = meta + (size_t)(n0 + 16 * i + lo) * (KCDIM/8) + 2 * hi;
  }
#pragma unroll
  for (int j = 0; j < 4; ++j)
    pb[j] = xh + (size_t)(m0 + 16 * j + lo) * KDIM + 16 * hi;

  for (int ks = 0; ks < KDIM / 64; ++ks) {
    const int kc0 = ks * 32;                       // compressed k offset
    const int k0  = ks * 64;                       // dense k offset

    // prefetch ~8 k-steps ahead on both streams (global_prefetch_b8)
    __builtin_prefetch(pa[0] + kc0 + 256, 0, 0);
    __builtin_prefetch(pb[0] + k0 + 512, 0, 0);

    v16h A[4];
    int  idx[4];
#pragma unroll
    for (int i = 0; i < 4; ++i) {
      A[i] = cat8(*(const v8h*)(pa[i] + kc0), *(const v8h*)(pa[i] + kc0 + 16));
      int2 mw = *(const int2*)(pm[i] + ks * 4);
      idx[i] = (mw.x & 0xFFFF) | (mw.y << 16);
    }
#pragma unroll
    for (int j = 0; j < 4; ++j) {
      v32h B = cat16(*(const v16h*)(pb[j] + k0), *(const v16h*)(pb[j] + k0 + 32));
#pragma unroll
      for (int i = 0; i < 4; ++i)
        acc[i][j] = SWMMAC_F16(A[i], B, acc[i][j], idx[i]);
    }
  }

  // D layout: VGPR r holds row (r + 8*hi), col = lo (per 16x16 tile)
#pragma unroll
  for (int i = 0; i < 4; ++i) {
#pragma unroll
    for (int j = 0; j < 4; ++j) {
#pragma unroll
      for (int r = 0; r < 8; ++r) {
        const size_t row = (size_t)(n0 + 16 * i + 8 * hi + r);
        out[row * MDIM + (m0 + 16 * j + lo)] = acc[i][j][r];
      }
    }
  }
}

extern "C" void kernel_launch(void* const* d_in, const int* in_sizes, int n_in,
                              void* d_out, int out_size, void* d_ws, size_t ws_size,
                              hipStream_t stream) {
  const float* x        = (const float*)d_in[0];
  const int*   qweight  = (const int*)d_in[1];
  const int*   g_idx    = (const int*)d_in[2];
  const int*   qzeros   = (const int*)d_in[3];
  const float* scales   = (const float*)d_in[4];
  const int*   meta     = (const int*)d_in[5];
  float*       out      = (float*)d_out;

  _Float16* xh = (_Float16*)d_ws;                                    // 16 MB
  _Float16* wh = (_Float16*)((char*)d_ws + (size_t)MDIM * KDIM * 2); // 45 MB

  xcvt_kernel<<<(MDIM * KDIM / 4) / 256, 256, 0, stream>>>(x, xh);
  dequant_kernel<<<dim3(NDIM / 32, KCDIM / 64), 256, 0, stream>>>(qweight, g_idx, qzeros,
                                                                  scales, wh);
  gemm_swmmac_kernel<<<dim3(NDIM / 64, MDIM / 64), 32, 0, stream>>>(wh, meta, xh, out);
}